// GraphAttentionLayer_47236050321919
// MI455X (gfx1250) — compile-verified
//
#include <hip/hip_runtime.h>
#include <math.h>

#define NEG_SLOPE 0.2f
#define LN_EPS 1e-5f

typedef __attribute__((ext_vector_type(2))) float v2f;
typedef __attribute__((ext_vector_type(8))) float v8f;

// ---------------------------------------------------------------------------
// Init: zero the aggregation region of d_out, write attention_weights = 1,
// init per-(node,head) max to -inf and denom to 0.
// ---------------------------------------------------------------------------
__global__ void gat_init(float* __restrict__ out_agg, float* __restrict__ attw,
                         float* __restrict__ mx, float* __restrict__ denom,
                         int N, int E) {
    int i = blockIdx.x * blockDim.x + threadIdx.x;
    if (i < N * 128) out_agg[i] = 0.0f;
    if (i < E)       attw[i] = 1.0f;
    if (i < N * 4) { mx[i] = -INFINITY; denom[i] = 0.0f; }
}

// ---------------------------------------------------------------------------
// GEMM: H = X(Nx128) @ W(128x128) via V_WMMA_F32_16X16X4_F32 (wave32).
// Block = (32,8): 8 waves cover the 8 N-tiles of one M-tile. The whole W
// (64KB) is staged into LDS once per block with async global->LDS B128 copies
// (ASYNCcnt + s_wait_asynccnt), then each wave reads its B fragments from LDS.
// A layout: lanes 0-15 -> M=lane, K={k,k+1}; lanes 16-31 -> M=lane-16, K={k+2,k+3}
// B layout (mirror): lanes 0-15 -> N=lane, K={k,k+1}; lanes 16-31 -> K={k+2,k+3}
// D layout: VGPR r, lanes 0-15 -> M=r, lanes 16-31 -> M=8+r; N=lane&15
// ---------------------------------------------------------------------------
__global__ void gat_gemm_wmma(const float* __restrict__ X,
                              const float* __restrict__ W,
                              float* __restrict__ Hh, int N) {
    __shared__ float Wlds[128 * 128];     // 64 KB of 320 KB WGP LDS

    const int lane  = threadIdx.x;        // 0..31
    const int tileN = threadIdx.y;        // 0..7
    const int tileM = blockIdx.x;
    const int half  = lane >> 4;          // 0 or 1
    const int l15   = lane & 15;

    // --- stage all of W into LDS: 256 threads x 16 async 16B copies ---
    {
        const int t = threadIdx.y * 32 + threadIdx.x;   // 0..255
        #pragma unroll
        for (int i = 0; i < 16; ++i) {
            const int idx = (t + i * 256) * 4;          // float index, 16B aligned
            const float* gp = W + idx;
            const unsigned ldsa = (unsigned)(size_t)(&Wlds[idx]);
            asm volatile("global_load_async_to_lds_b128 %0, %1, off"
                         :: "v"(ldsa), "v"(gp)
                         : "memory");
        }
        asm volatile("s_wait_asynccnt 0x0" ::: "memory");
    }
    __syncthreads();

    int m = tileM * 16 + l15;
    if (m >= N) m = N - 1;                // clamp (EXEC must stay all-ones)
    const int n = tileN * 16 + l15;

    const float* xrow = X + (size_t)m * 128;
    v8f acc = {};

    #pragma unroll
    for (int k = 0; k < 128; k += 4) {
        const int ka = k + 2 * half;
        v2f a = *(const v2f*)(xrow + ka);
        v2f b;
        b.x = Wlds[ka * 128 + n];
        b.y = Wlds[(ka + 1) * 128 + n];
        acc = __builtin_amdgcn_wmma_f32_16x16x4_f32(
            /*neg_a=*/false, a, /*neg_b=*/false, b,
            /*c_mod=*/(short)0, acc, /*reuse_a=*/false, /*reuse_b=*/false);
    }

    const int rowBase = tileM * 16 + 8 * half;
    #pragma unroll
    for (int r = 0; r < 8; ++r) {
        const int row = rowBase + r;
        if (row < N) Hh[(size_t)row * 128 + tileN * 16 + l15] = acc[r];
    }
}

// ---------------------------------------------------------------------------
// Per-node attention logits: a_src[n,h] = sum_d Hh[n,h,d]*att_src[h,d] (same
// for dst). One wave per node; lane covers 4 contiguous channels (head = lane/8).
// ---------------------------------------------------------------------------
__global__ void gat_att(const float* __restrict__ Hh,
                        const float* __restrict__ att_src,
                        const float* __restrict__ att_dst,
                        float* __restrict__ a_src, float* __restrict__ a_dst,
                        int N) {
    const int wave = (blockIdx.x * blockDim.x + threadIdx.x) >> 5;
    const int lane = threadIdx.x & 31;
    if (wave >= N) return;

    const float4 hv = *(const float4*)(Hh + (size_t)wave * 128 + lane * 4);
    const float4 as = *(const float4*)(att_src + lane * 4);
    const float4 ad = *(const float4*)(att_dst + lane * 4);
    float ps = hv.x * as.x + hv.y * as.y + hv.z * as.z + hv.w * as.w;
    float pd = hv.x * ad.x + hv.y * ad.y + hv.z * ad.z + hv.w * ad.w;
    // reduce within each 8-lane head group
    #pragma unroll
    for (int off = 4; off >= 1; off >>= 1) {
        ps += __shfl_xor(ps, off);
        pd += __shfl_xor(pd, off);
    }
    if ((lane & 7) == 0) {
        const int head = lane >> 3;
        a_src[wave * 4 + head] = ps;
        a_dst[wave * 4 + head] = pd;
    }
}

__device__ __forceinline__ float leaky(float e) {
    return e > 0.0f ? e : NEG_SLOPE * e;
}

// Float atomic max via int bit tricks (valid with -inf init).
__device__ __forceinline__ void atomic_max_f32(float* addr, float val) {
    if (val >= 0.0f)
        atomicMax((int*)addr, __float_as_int(val));
    else
        atomicMin((unsigned int*)addr, __float_as_uint(val));
}

// ---------------------------------------------------------------------------
// Edge pass 1: segment max of leaky_relu(a_src[s]+a_dst[d]) per dst node.
// Indices i in [E, E+N) are the implicit self-loops.
// ---------------------------------------------------------------------------
__global__ void gat_edge_max(const int* __restrict__ ei,
                             const float* __restrict__ a_src,
                             const float* __restrict__ a_dst,
                             float* __restrict__ mx, int N, int E) {
    const int i = blockIdx.x * blockDim.x + threadIdx.x;
    if (i >= E + N) return;
    int s, d;
    if (i < E) { s = ei[i]; d = ei[E + i]; } else { s = d = i - E; }
    const float4 asv = *(const float4*)(a_src + (size_t)s * 4);
    const float4 adv = *(const float4*)(a_dst + (size_t)d * 4);
    atomic_max_f32(&mx[d * 4 + 0], leaky(asv.x + adv.x));
    atomic_max_f32(&mx[d * 4 + 1], leaky(asv.y + adv.y));
    atomic_max_f32(&mx[d * 4 + 2], leaky(asv.z + adv.z));
    atomic_max_f32(&mx[d * 4 + 3], leaky(asv.w + adv.w));
}

// ---------------------------------------------------------------------------
// Edge pass 2: denom[d,h] = sum exp(e - max)
// ---------------------------------------------------------------------------
__global__ void gat_edge_sum(const int* __restrict__ ei,
                             const float* __restrict__ a_src,
                             const float* __restrict__ a_dst,
                             const float* __restrict__ mx,
                             float* __restrict__ denom, int N, int E) {
    const int i = blockIdx.x * blockDim.x + threadIdx.x;
    if (i >= E + N) return;
    int s, d;
    if (i < E) { s = ei[i]; d = ei[E + i]; } else { s = d = i - E; }
    const float4 asv = *(const float4*)(a_src + (size_t)s * 4);
    const float4 adv = *(const float4*)(a_dst + (size_t)d * 4);
    const float4 mxv = *(const float4*)(mx + (size_t)d * 4);
    atomicAdd(&denom[d * 4 + 0], __expf(leaky(asv.x + adv.x) - mxv.x));
    atomicAdd(&denom[d * 4 + 1], __expf(leaky(asv.y + adv.y) - mxv.y));
    atomicAdd(&denom[d * 4 + 2], __expf(leaky(asv.z + adv.z) - mxv.z));
    atomicAdd(&denom[d * 4 + 3], __expf(leaky(asv.w + adv.w) - mxv.w));
}

// ---------------------------------------------------------------------------
// Edge pass 3: out[d] += alpha * Hh[s]. One wave per edge; lane owns 4
// contiguous channels (all within one head), gathers 16B and does 4 atomic
// fp32 adds.
// ---------------------------------------------------------------------------
__global__ void gat_edge_agg(const int* __restrict__ ei,
                             const float* __restrict__ Hh,
                             const float* __restrict__ a_src,
                             const float* __restrict__ a_dst,
                             const float* __restrict__ mx,
                             const float* __restrict__ denom,
                             float* __restrict__ out, int N, int E) {
    const int wave = (blockIdx.x * blockDim.x + threadIdx.x) >> 5;
    const int lane = threadIdx.x & 31;
    if (wave >= E + N) return;
    int s, d;
    if (wave < E) { s = ei[wave]; d = ei[E + wave]; } else { s = d = wave - E; }

    const int head = lane >> 3;
    const float e = leaky(a_src[s * 4 + head] + a_dst[d * 4 + head]);
    const float alpha = __expf(e - mx[d * 4 + head]) / (denom[d * 4 + head] + 1e-16f);

    const float4 hv = *(const float4*)(Hh + (size_t)s * 128 + lane * 4);
    float* o = out + (size_t)d * 128 + lane * 4;
    atomicAdd(o + 0, hv.x * alpha);
    atomicAdd(o + 1, hv.y * alpha);
    atomicAdd(o + 2, hv.z * alpha);
    atomicAdd(o + 3, hv.w * alpha);
}

// ---------------------------------------------------------------------------
// Finalize: out = LN(agg + bias + x) * gamma + beta, in place. Wave per node.
// ---------------------------------------------------------------------------
__global__ void gat_finalize(const float* __restrict__ X,
                             const float* __restrict__ bias,
                             const float* __restrict__ gamma,
                             const float* __restrict__ beta,
                             float* __restrict__ out, int N) {
    const int wave = (blockIdx.x * blockDim.x + threadIdx.x) >> 5;
    const int lane = threadIdx.x & 31;
    if (wave >= N) return;

    float* orow = out + (size_t)wave * 128;
    float4 v  = *(const float4*)(orow + lane * 4);
    const float4 xv = *(const float4*)(X + (size_t)wave * 128 + lane * 4);
    const float4 bv = *(const float4*)(bias + lane * 4);
    v.x += xv.x + bv.x; v.y += xv.y + bv.y;
    v.z += xv.z + bv.z; v.w += xv.w + bv.w;

    float sum = v.x + v.y + v.z + v.w;
    float sq  = v.x * v.x + v.y * v.y + v.z * v.z + v.w * v.w;
    #pragma unroll
    for (int off = 16; off >= 1; off >>= 1) {
        sum += __shfl_xor(sum, off);
        sq  += __shfl_xor(sq, off);
    }
    const float mean = sum * (1.0f / 128.0f);
    const float var  = sq * (1.0f / 128.0f) - mean * mean;
    const float rstd = rsqrtf(var + LN_EPS);

    const float4 g  = *(const float4*)(gamma + lane * 4);
    const float4 be = *(const float4*)(beta + lane * 4);
    v.x = (v.x - mean) * rstd * g.x + be.x;
    v.y = (v.y - mean) * rstd * g.y + be.y;
    v.z = (v.z - mean) * rstd * g.z + be.z;
    v.w = (v.w - mean) * rstd * g.w + be.w;
    *(float4*)(orow + lane * 4) = v;
}

// ---------------------------------------------------------------------------
extern "C" void kernel_launch(void* const* d_in, const int* in_sizes, int n_in,
                              void* d_out, int out_size, void* d_ws, size_t ws_size,
                              hipStream_t stream) {
    const float* x       = (const float*)d_in[0];
    const int*   ei      = (const int*)d_in[1];   // [2,E]: row0=src, row1=dst
    const float* W       = (const float*)d_in[2];
    const float* att_src = (const float*)d_in[3];
    const float* att_dst = (const float*)d_in[4];
    const float* bias    = (const float*)d_in[5];
    const float* gamma   = (const float*)d_in[6];
    const float* beta    = (const float*)d_in[7];

    const int N = in_sizes[0] / 128;
    const int E = in_sizes[1] / 2;

    float* out  = (float*)d_out;                  // [N*128] aggregate -> final
    float* attw = out + (size_t)N * 128;          // [E] ones

    float* Hh    = (float*)d_ws;                  // [N*128]
    float* a_src = Hh + (size_t)N * 128;          // [N*4]
    float* a_dst = a_src + (size_t)N * 4;         // [N*4]
    float* mx    = a_dst + (size_t)N * 4;         // [N*4]
    float* denom = mx + (size_t)N * 4;            // [N*4]

    const int totalE = E + N;

    gat_init<<<(N * 128 + 255) / 256, 256, 0, stream>>>(out, attw, mx, denom, N, E);

    gat_gemm_wmma<<<(N + 15) / 16, dim3(32, 8), 0, stream>>>(x, W, Hh, N);

    gat_att<<<(N * 32 + 255) / 256, 256, 0, stream>>>(Hh, att_src, att_dst,
                                                      a_src, a_dst, N);

    gat_edge_max<<<(totalE + 255) / 256, 256, 0, stream>>>(ei, a_src, a_dst, mx, N, E);

    gat_edge_sum<<<(totalE + 255) / 256, 256, 0, stream>>>(ei, a_src, a_dst, mx,
                                                           denom, N, E);

    gat_edge_agg<<<((size_t)totalE * 32 + 255) / 256, 256, 0, stream>>>(
        ei, Hh, a_src, a_dst, mx, denom, out, N, E);

    gat_finalize<<<(N * 32 + 255) / 256, 256, 0, stream>>>(x, bias, gamma, beta,
                                                           out, N);
}